// RNNLayer_27075473834256
// MI455X (gfx1250) — compile-verified
//
#include <hip/hip_runtime.h>
#include <hip/hip_bf16.h>
#include <math.h>

typedef __attribute__((ext_vector_type(2))) float v2f;
typedef __attribute__((ext_vector_type(4))) float v4f;
typedef __attribute__((ext_vector_type(8))) float v8f;

#define UNITS   1024
#define OUT_DIM 10
#define FT_DIM  128
#define BATCH   256
#define TSTEPS  256

// GEMM tiling: block tile 32(M) x 128(N), K-chunks of 32, double-buffered LDS.
// 128 threads = 4 waves; wave grid 2(M) x 2(N); each wave: 16x64 = 4 WMMA C tiles.
#define BM 32
#define BN 128
#define BK 32
#define LDA  36   // As row stride: mult-of-4 (16B-aligned v4 stores), even (b64 frags), conflict-free
#define LDBT 34   // Bt row stride (transposed [n][k]): even, both lane halves hit disjoint banks
#define A_SZ (BM * LDA)   // 1152 floats
#define B_SZ (BN * LDBT)  // 4352 floats

// AMODE: 0 = proj (A = inputs [B,T,F], M row r -> input row (r&255)*T + (r>>8))
//        1 = step (A = Xprev [BATCH, UNITS] row-major)
//        2 = step t==0 (A = x0 [UNITS], broadcast to every row)
// EPI_TANH: false -> C[m,n] = acc ; true -> C[m,n] = tanh(acc + C[m,n] + bias[n])
template <int KDIM, int AMODE, bool EPI_TANH>
__global__ __launch_bounds__(128) void gemm_kernel(const float* __restrict__ A,
                                                   const float* __restrict__ B,
                                                   const float* __restrict__ bias,
                                                   float* __restrict__ C) {
    __shared__ float As[2 * A_SZ];
    __shared__ float Bt[2 * B_SZ];

    const int bn0 = blockIdx.x * BN;
    const int bm0 = blockIdx.y * BM;
    const int tid = threadIdx.x;
    const int lane = tid & 31, wid = tid >> 5;
    const int wm = wid & 1, wn = wid >> 1;
    const int hi = lane >> 4, l16 = lane & 15;

    v8f acc[4] = {};
    v4f aReg[2];   // global->reg staging: A 32x32 / 128 thr = 2 vec4
    v4f bReg[8];   // B 32x128 / 128 thr = 8 vec4

    // ---- global -> registers (chunk at kc) ----
    auto gload = [&](int kc) {
#pragma unroll
        for (int i = 0; i < 2; ++i) {
            const int v = tid + 128 * i;
            const int row = v >> 3, c4 = (v & 7) * 4;
            size_t off;
            if (AMODE == 0) {
                const int r = bm0 + row;
                off = (size_t)((r & 255) * TSTEPS + (r >> 8)) * FT_DIM + kc + c4;
            } else if (AMODE == 1) {
                off = (size_t)(bm0 + row) * UNITS + kc + c4;
            } else {
                off = (size_t)(kc + c4);
            }
            aReg[i] = *(const v4f*)(A + off);
        }
        // lanes sweep k (v&31) so the transpose stores below are conflict-free
#pragma unroll
        for (int i = 0; i < 8; ++i) {
            const int v = tid + 128 * i;
            const int k = v & 31, ng = v >> 5;
            bReg[i] = *(const v4f*)(B + (size_t)(kc + k) * UNITS + bn0 + ng * 4);
        }
    };

    // ---- registers -> LDS (buffer sel) ----
    auto sstore = [&](int buf) {
        float* As_ = As + buf * A_SZ;
        float* Bt_ = Bt + buf * B_SZ;
#pragma unroll
        for (int i = 0; i < 2; ++i) {
            const int v = tid + 128 * i;
            const int row = v >> 3, c4 = (v & 7) * 4;
            *(v4f*)&As_[row * LDA + c4] = aReg[i];
        }
#pragma unroll
        for (int i = 0; i < 8; ++i) {
            const int v = tid + 128 * i;
            const int k = v & 31, ng = v >> 5;
#pragma unroll
            for (int j = 0; j < 4; ++j)
                Bt_[(ng * 4 + j) * LDBT + k] = bReg[i][j];
        }
    };

    gload(0);
    sstore(0);
    __syncthreads();

    int buf = 0;
    for (int kc = 0; kc < KDIM; kc += BK) {
        const bool more = (kc + BK) < KDIM;
        if (more) gload(kc + BK);   // prefetch next chunk; overlaps with WMMA below

        const float* As_ = As + buf * A_SZ;
        const float* Bt_ = Bt + buf * B_SZ;
        const int mrow = wm * 16 + l16;
#pragma unroll
        for (int kk = 0; kk < BK; kk += 4) {
            const int ka = kk + 2 * hi;          // lanes 0-15: K0,K1 ; lanes 16-31: K2,K3
            v2f a = *(const v2f*)&As_[mrow * LDA + ka];
#pragma unroll
            for (int tt = 0; tt < 4; ++tt) {
                const int n = wn * 64 + tt * 16 + l16;
                v2f b = *(const v2f*)&Bt_[n * LDBT + ka];  // single b64, even VGPR pair
                acc[tt] = __builtin_amdgcn_wmma_f32_16x16x4_f32(
                    false, a, false, b, (short)0, acc[tt], false, false);
            }
        }

        if (more) sstore(buf ^ 1);
        __syncthreads();
        buf ^= 1;
    }

#pragma unroll
    for (int tt = 0; tt < 4; ++tt) {
        const int n = bn0 + wn * 64 + tt * 16 + l16;
        float bn = 0.f;
        if (EPI_TANH) bn = bias[n];
#pragma unroll
        for (int i = 0; i < 8; ++i) {
            const int m = bm0 + wm * 16 + i + 8 * hi;
            const size_t idx = (size_t)m * UNITS + n;
            if (EPI_TANH)
                C[idx] = tanhf(acc[tt][i] + C[idx] + bn);
            else
                C[idx] = acc[tt][i];
        }
    }
}

// ---------------------------------------------------------------------------
// Output head: y = x_last @ W_out + bias_out ; softmax. One wave per batch row.
// ---------------------------------------------------------------------------
__global__ __launch_bounds__(32) void head_kernel(const float* __restrict__ Xlast,
                                                  const float* __restrict__ Wo,
                                                  const float* __restrict__ bo,
                                                  float* __restrict__ Y) {
    const int b = blockIdx.x;
    const int lane = threadIdx.x;
    float acc[OUT_DIM];
#pragma unroll
    for (int j = 0; j < OUT_DIM; ++j) acc[j] = 0.f;

    const float* xr = Xlast + (size_t)b * UNITS;
    for (int k = lane; k < UNITS; k += 32) {
        const float xv = xr[k];
#pragma unroll
        for (int j = 0; j < OUT_DIM; ++j)
            acc[j] = fmaf(xv, Wo[(size_t)k * OUT_DIM + j], acc[j]);
    }
#pragma unroll
    for (int j = 0; j < OUT_DIM; ++j) {
#pragma unroll
        for (int off = 16; off; off >>= 1) acc[j] += __shfl_xor(acc[j], off, 32);
        acc[j] += bo[j];
    }
    float mx = acc[0];
#pragma unroll
    for (int j = 1; j < OUT_DIM; ++j) mx = fmaxf(mx, acc[j]);
    float s = 0.f;
#pragma unroll
    for (int j = 0; j < OUT_DIM; ++j) { acc[j] = expf(acc[j] - mx); s += acc[j]; }
    const float inv = 1.0f / s;
    if (lane < OUT_DIM) Y[(size_t)b * OUT_DIM + lane] = acc[lane] * inv;
}

// ---------------------------------------------------------------------------
extern "C" void kernel_launch(void* const* d_in, const int* in_sizes, int n_in,
                              void* d_out, int out_size, void* d_ws, size_t ws_size,
                              hipStream_t stream) {
    const float* inputs   = (const float*)d_in[0];
    const float* V        = (const float*)d_in[1];
    const float* W        = (const float*)d_in[2];
    const float* bias     = (const float*)d_in[3];
    const float* W_out    = (const float*)d_in[4];
    const float* bias_out = (const float*)d_in[5];
    const float* x0       = (const float*)d_in[6];

    float* Xall = (float*)d_out;                              // [T, B, U]
    float* Y    = Xall + (size_t)TSTEPS * BATCH * UNITS;      // [B, OUT_DIM]

    dim3 blk(128);

    // h_all = inputs @ V, written straight into d_out (X_all region, row r = t*B+b)
    dim3 gproj(UNITS / BN, (TSTEPS * BATCH) / BM);            // 8 x 2048
    gemm_kernel<FT_DIM, 0, false><<<gproj, blk, 0, stream>>>(inputs, V, nullptr, Xall);

    // sequential scan: one launch per timestep, in place on d_out[t]
    dim3 gstep(UNITS / BN, BATCH / BM);                       // 8 x 8
    gemm_kernel<UNITS, 2, true><<<gstep, blk, 0, stream>>>(x0, W, bias, Xall);
    for (int t = 1; t < TSTEPS; ++t) {
        gemm_kernel<UNITS, 1, true><<<gstep, blk, 0, stream>>>(
            Xall + (size_t)(t - 1) * BATCH * UNITS, W, bias,
            Xall + (size_t)t * BATCH * UNITS);
    }

    // output head + softmax
    head_kernel<<<BATCH, 32, 0, stream>>>(
        Xall + (size_t)(TSTEPS - 1) * BATCH * UNITS, W_out, bias_out, Y);
}